// BaseGraphModel_1872605741079
// MI455X (gfx1250) — compile-verified
//
#include <hip/hip_runtime.h>
#include <hip/hip_bf16.h>

typedef float v2f __attribute__((ext_vector_type(2)));
typedef float v8f __attribute__((ext_vector_type(8)));
typedef int   v4i __attribute__((vector_size(16)));

#define N_NODES   50000
#define N_EDGES   800000
#define NUM_GRAPHS 128
#define DHID      256
#define NUM_CLASSES 40
#define BN_EPS    1e-5f

#if defined(__has_builtin)
#if __has_builtin(__builtin_amdgcn_global_load_async_to_lds_b128) && \
    __has_builtin(__builtin_amdgcn_s_wait_asynccnt)
#define HAVE_ASYNC_LDS 1
#endif
#endif
#ifndef HAVE_ASYNC_LDS
#define HAVE_ASYNC_LDS 0
#endif

// ---------------------------------------------------------------------------
// Generic fill
// ---------------------------------------------------------------------------
__global__ void fill_kernel(float* __restrict__ p, float v, int n) {
  int i = blockIdx.x * blockDim.x + threadIdx.x;
  if (i < n) p[i] = v;
}

// ---------------------------------------------------------------------------
// Degree: deg[dst] += 1 per edge (deg pre-filled with 1.0 for self loop),
// then in-place dinv = rsqrt(max(deg,1))
// ---------------------------------------------------------------------------
__global__ void deg_kernel(const long long* __restrict__ ei, float* __restrict__ deg) {
  int e = blockIdx.x * blockDim.x + threadIdx.x;
  if (e < N_EDGES) unsafeAtomicAdd(&deg[(int)ei[N_EDGES + e]], 1.0f);
}

__global__ void dinv_kernel(float* __restrict__ deg) {
  int i = blockIdx.x * blockDim.x + threadIdx.x;
  if (i < N_NODES) deg[i] = rsqrtf(fmaxf(deg[i], 1.0f));
}

// ---------------------------------------------------------------------------
// GEMM H[N x 256] = X[N x K] * W[K x 256] via V_WMMA_F32_16X16X4_F32.
// 128 threads = 4 waves; block -> one 16-row tile; wave w -> cols [64w, 64w+64).
// The 16 x K A-tile is staged once in LDS (async-to-LDS on CDNA5), removing the
// 4x redundant global reads of X across the block's waves. LDS row pitch K+4
// floats keeps the A-fragment ds_load_b64s bank-conflict-free.
// N_NODES = 50000 = 3125 * 16 exactly, so no row guards needed.
// ---------------------------------------------------------------------------
__global__ __launch_bounds__(128) void gemm_wmma_kernel(
    const float* __restrict__ X, const float* __restrict__ W,
    float* __restrict__ H, int K) {
  __shared__ float xs[16 * (DHID + 4)];
  const int Kp   = K + 4;
  const int tid  = threadIdx.x;
  const int row0 = blockIdx.x * 16;

  // ---- stage A tile: 16 rows x K floats, 128 threads x float4 chunks ----
  const int vecPerRow = K >> 2;        // 32 or 64
  const int nvec      = 16 * vecPerRow;
  for (int v = tid; v < nvec; v += 128) {
    int r  = v / vecPerRow;
    int cv = (v - r * vecPerRow) << 2;
    const float* g = X + (size_t)(row0 + r) * K + cv;
    float* l = &xs[r * Kp + cv];
#if HAVE_ASYNC_LDS
    __builtin_amdgcn_global_load_async_to_lds_b128(
        (__attribute__((address_space(1))) v4i*)g,
        (__attribute__((address_space(3))) v4i*)l, 0, 0);
#else
    *(float4*)l = *(const float4*)g;
#endif
  }
#if HAVE_ASYNC_LDS
  __builtin_amdgcn_s_wait_asynccnt(0);
#endif
  __syncthreads();

  // ---- WMMA main loop ----
  const int lane = tid & 31;
  const int wave = tid >> 5;
  const int col0 = wave * 64;
  const int m    = lane & 15;
  const int kh   = (lane >> 4) * 2;    // 0 for lanes 0-15, 2 for lanes 16-31

  v8f acc0 = {}, acc1 = {}, acc2 = {}, acc3 = {};
  const float* xr = &xs[m * Kp + kh];

  for (int kk = 0; kk < K; kk += 4) {
    v2f a;
    a.x = xr[kk];
    a.y = xr[kk + 1];
    const float* w0 = W + (size_t)(kk + kh) * DHID + col0 + m;
    const float* w1 = w0 + DHID;
    v2f b0, b1, b2, b3;
    b0.x = w0[0];  b0.y = w1[0];
    b1.x = w0[16]; b1.y = w1[16];
    b2.x = w0[32]; b2.y = w1[32];
    b3.x = w0[48]; b3.y = w1[48];
    acc0 = __builtin_amdgcn_wmma_f32_16x16x4_f32(false, a, false, b0, (short)0, acc0, false, false);
    acc1 = __builtin_amdgcn_wmma_f32_16x16x4_f32(false, a, false, b1, (short)0, acc1, false, false);
    acc2 = __builtin_amdgcn_wmma_f32_16x16x4_f32(false, a, false, b2, (short)0, acc2, false, false);
    acc3 = __builtin_amdgcn_wmma_f32_16x16x4_f32(false, a, false, b3, (short)0, acc3, false, false);
  }

  // C/D layout: VGPR r -> row (r + 8*(lane>=16)), col = lane&15
  const int n    = lane & 15;
  const int mrow = (lane >> 4) * 8;
  float* out = H + (size_t)(row0 + mrow) * DHID + col0 + n;
#pragma unroll
  for (int r = 0; r < 8; ++r) {
    out[(size_t)r * DHID +  0] = acc0[r];
    out[(size_t)r * DHID + 16] = acc1[r];
    out[(size_t)r * DHID + 32] = acc2[r];
    out[(size_t)r * DHID + 48] = acc3[r];
  }
}

// ---------------------------------------------------------------------------
// Self-loop initializer: AGG[n][c] = H[n][c] * dinv[n]^2  (also zero-inits AGG,
// overwriting the previous layer's dead X)
// ---------------------------------------------------------------------------
__global__ void selfloop_kernel(const float* __restrict__ H, const float* __restrict__ dinv,
                                float* __restrict__ AGG) {
  size_t i = (size_t)blockIdx.x * blockDim.x + threadIdx.x;
  if (i >= (size_t)N_NODES * DHID) return;
  int node = (int)(i >> 8);
  float c = dinv[node];
  AGG[i] = H[i] * c * c;
}

// ---------------------------------------------------------------------------
// Edge aggregation: one wave per edge; lane l handles cols l, l+32, ... (coalesced)
// AGG[dst][c] += H[src][c] * dinv[src]*dinv[dst]
// ---------------------------------------------------------------------------
__global__ __launch_bounds__(256) void edge_kernel(
    const long long* __restrict__ ei, const float* __restrict__ dinv,
    const float* __restrict__ H, float* __restrict__ AGG) {
  int lane = threadIdx.x & 31;
  int e = (blockIdx.x << 3) + (threadIdx.x >> 5);
  if (e >= N_EDGES) return;
  int s = (int)ei[e];
  int d = (int)ei[N_EDGES + e];
  float c = dinv[s] * dinv[d];
  const float* hs = H + (size_t)s * DHID;
  float* ad = AGG + (size_t)d * DHID;
#pragma unroll
  for (int j = 0; j < 8; ++j) {
    int col = lane + (j << 5);
    unsafeAtomicAdd(&ad[col], hs[col] * c);
  }
}

// ---------------------------------------------------------------------------
// BatchNorm stats: stats[col] = sum, stats[256+col] = sumsq
// ---------------------------------------------------------------------------
__global__ __launch_bounds__(256) void bnstats_kernel(const float* __restrict__ AGG,
                                                      float* __restrict__ stats) {
  int col = threadIdx.x;
  float s = 0.f, s2 = 0.f;
  for (int r = blockIdx.x; r < N_NODES; r += gridDim.x) {
    float v = AGG[(size_t)r * DHID + col];
    s += v; s2 += v * v;
  }
  unsafeAtomicAdd(&stats[col], s);
  unsafeAtomicAdd(&stats[DHID + col], s2);
}

// ---------------------------------------------------------------------------
// BN apply + ReLU, IN PLACE on AGG: h = relu((h-mean)*gamma*rsqrt(var+eps)+beta)
// (conv bias cancels exactly inside BN, so it is skipped)
// ---------------------------------------------------------------------------
__global__ void bnapply_kernel(float* __restrict__ AGG, const float* __restrict__ stats,
                               const float* __restrict__ gamma, const float* __restrict__ beta) {
  size_t i = (size_t)blockIdx.x * blockDim.x + threadIdx.x;
  if (i >= (size_t)N_NODES * DHID) return;
  int col = (int)(i & (DHID - 1));
  const float invN = 1.0f / (float)N_NODES;
  float mean = stats[col] * invN;
  float var  = stats[DHID + col] * invN - mean * mean;
  float scale = gamma[col] * rsqrtf(var + BN_EPS);
  float v = (AGG[i] - mean) * scale + beta[col];
  AGG[i] = fmaxf(v, 0.0f);
}

// ---------------------------------------------------------------------------
// Global mean pool: sums + counts (atomics)
// ---------------------------------------------------------------------------
__global__ void pool_kernel(const float* __restrict__ Xf, const long long* __restrict__ batch,
                            float* __restrict__ pool) {
  size_t i = (size_t)blockIdx.x * blockDim.x + threadIdx.x;
  if (i >= (size_t)N_NODES * DHID) return;
  int node = (int)(i >> 8);
  int col  = (int)(i & (DHID - 1));
  int g = (int)batch[node];
  unsafeAtomicAdd(&pool[(size_t)g * DHID + col], Xf[i]);
}

__global__ void cnt_kernel(const long long* __restrict__ batch, float* __restrict__ cnt) {
  int n = blockIdx.x * blockDim.x + threadIdx.x;
  if (n < N_NODES) unsafeAtomicAdd(&cnt[(int)batch[n]], 1.0f);
}

// ---------------------------------------------------------------------------
// Classifier: out[g][k] = (pool[g]/max(cnt,1)) . W_lin[:,k] + b_lin[k]
// ---------------------------------------------------------------------------
__global__ void classifier_kernel(const float* __restrict__ pool, const float* __restrict__ cnt,
                                  const float* __restrict__ Wl, const float* __restrict__ bl,
                                  float* __restrict__ out) {
  int i = blockIdx.x * blockDim.x + threadIdx.x;
  if (i >= NUM_GRAPHS * NUM_CLASSES) return;
  int g = i / NUM_CLASSES, k = i % NUM_CLASSES;
  float inv = 1.0f / fmaxf(cnt[g], 1.0f);
  const float* p = pool + (size_t)g * DHID;
  float acc = bl[k];
  for (int c = 0; c < DHID; ++c) acc += p[c] * inv * Wl[c * NUM_CLASSES + k];
  out[i] = acc;
}

// ---------------------------------------------------------------------------
extern "C" void kernel_launch(void* const* d_in, const int* in_sizes, int n_in,
                              void* d_out, int out_size, void* d_ws, size_t ws_size,
                              hipStream_t stream) {
  const float*     x     = (const float*)d_in[0];
  const long long* ei    = (const long long*)d_in[1];
  const long long* batch = (const long long*)d_in[2];
  const float* W[3]     = { (const float*)d_in[3],  (const float*)d_in[7],  (const float*)d_in[11] };
  const float* gamma[3] = { (const float*)d_in[5],  (const float*)d_in[9],  (const float*)d_in[13] };
  const float* beta[3]  = { (const float*)d_in[6],  (const float*)d_in[10], (const float*)d_in[14] };
  const float* W_lin = (const float*)d_in[15];
  const float* b_lin = (const float*)d_in[16];
  float* out = (float*)d_out;

  // workspace layout (floats)
  float* ws = (float*)d_ws;
  const size_t NM = (size_t)N_NODES * DHID;        // 12.8M
  float* BUF_A = ws;                               // GEMM out (H)
  float* BUF_B = BUF_A + NM;                       // AGG / BN output / next X
  float* dinv  = BUF_B + NM;                       // N_NODES
  float* stats = dinv + N_NODES;                   // 512
  float* pool  = stats + 2 * DHID;                 // 128*256
  float* cnt   = pool + (size_t)NUM_GRAPHS * DHID; // 128

  const int T = 256;
  const int elemBlocks = (int)((NM + T - 1) / T);

  // degrees -> dinv (self loop contributes the initial 1.0)
  fill_kernel<<<(N_NODES + T - 1) / T, T, 0, stream>>>(dinv, 1.0f, N_NODES);
  deg_kernel<<<(N_EDGES + T - 1) / T, T, 0, stream>>>(ei, dinv);
  dinv_kernel<<<(N_NODES + T - 1) / T, T, 0, stream>>>(dinv);

  const float* Xcur = x;
  int K = 128;
  for (int layer = 0; layer < 3; ++layer) {
    // H = X @ W   (50000/16 = 3125 row tiles, 4 waves cover all 256 cols)
    gemm_wmma_kernel<<<N_NODES / 16, 128, 0, stream>>>(Xcur, W[layer], BUF_A, K);
    // AGG(BUF_B) = selfloop init, then scatter-add over edges
    selfloop_kernel<<<elemBlocks, T, 0, stream>>>(BUF_A, dinv, BUF_B);
    edge_kernel<<<(N_EDGES + 7) / 8, 256, 0, stream>>>(ei, dinv, BUF_A, BUF_B);
    // BatchNorm + ReLU (in place on BUF_B)
    fill_kernel<<<2, T, 0, stream>>>(stats, 0.0f, 2 * DHID);
    bnstats_kernel<<<512, 256, 0, stream>>>(BUF_B, stats);
    bnapply_kernel<<<elemBlocks, T, 0, stream>>>(BUF_B, stats, gamma[layer], beta[layer]);
    Xcur = BUF_B;
    K = DHID;
  }

  // mean pool + classifier
  fill_kernel<<<(NUM_GRAPHS * DHID + NUM_GRAPHS + T - 1) / T, T, 0, stream>>>(
      pool, 0.0f, NUM_GRAPHS * DHID + NUM_GRAPHS);
  pool_kernel<<<elemBlocks, T, 0, stream>>>(Xcur, batch, pool);
  cnt_kernel<<<(N_NODES + T - 1) / T, T, 0, stream>>>(batch, cnt);
  classifier_kernel<<<(NUM_GRAPHS * NUM_CLASSES + T - 1) / T, T, 0, stream>>>(
      pool, cnt, W_lin, b_lin, out);
}